// lg2graphNode_21663815041154
// MI455X (gfx1250) — compile-verified
//
#include <hip/hip_runtime.h>
#include <stdint.h>

// ---------------------------------------------------------------------------
// lg2graphNode (lgvariant 21) for MI455X / gfx1250
//
// out[:, 0:256]   = segment_mean(x[:, 0:256],  gidx1)   (incoming)
// out[:, 256:384] = segment_mean(x[:, 256:384], gidx0)  (outgoing)
//
// Memory-bound scatter (no matmul -> WMMA not applicable).
// CDNA5 paths used:
//   * global_load_async_to_lds_b128 (ASYNCcnt) with th:TH_LOAD_NT:
//     - 12 x 16B/lane in flight with zero VGPR cost
//     - NT keeps the streamed 491MB of x from evicting the 245MB atomic
//       accumulator out of the 192MB L2 (atomics are L2-side RMWs)
//   * progressive s_wait_asynccnt pipelining (consume chunk i while
//     chunks i+1..11 are still in flight)
//   * global_atomic_add_f32, non-returning (STOREcnt)
// ---------------------------------------------------------------------------

#define D_DIM      384
#define COLS4      96          // 384 / 4
#define SPLIT_COL  256         // 2*H
#define CHUNKS_PB  3072        // 32 edges * 96 float4-chunks, per block
#define STAGE_IT   12          // 3072 / 256 threads

// ---------------- zero kernels ----------------
__global__ void zero_f4_kernel(float4* __restrict__ p, long n4) {
    long i = (long)blockIdx.x * blockDim.x + threadIdx.x;
    if (i < n4) p[i] = make_float4(0.f, 0.f, 0.f, 0.f);
}

__global__ void zero_f_kernel(float* __restrict__ p, long n) {
    long i = (long)blockIdx.x * blockDim.x + threadIdx.x;
    if (i < n) p[i] = 0.f;
}

// ---------------- per-graph node offsets (cumsum, B is tiny) ----------------
__global__ void offsets_kernel(const int* __restrict__ gsz, int B,
                               int* __restrict__ offs) {
    if (blockIdx.x == 0 && threadIdx.x == 0) {
        int acc = 0;
        for (int b = 0; b < B; ++b) { offs[b] = acc; acc += gsz[b]; }
    }
}

// ---------------- global edge indices + endpoint counts ----------------
__global__ void gidx_count_kernel(const int* __restrict__ lg,
                                  const int* __restrict__ ptr,
                                  const int* __restrict__ offs,
                                  int B, long E,
                                  int2* __restrict__ gidx,
                                  float* __restrict__ cnt_in,
                                  float* __restrict__ cnt_out) {
    long e = (long)blockIdx.x * blockDim.x + threadIdx.x;
    if (e >= E) return;
    int b = 0;
    while (b + 1 < B && e >= (long)ptr[b + 1]) ++b;   // B=16: short cached scan
    int off = offs[b];
    int g0 = lg[2 * e]     + off;   // outgoing endpoint (idx[:,0])
    int g1 = lg[2 * e + 1] + off;   // incoming endpoint (idx[:,1])
    int2 g; g.x = g0; g.y = g1;
    gidx[e] = g;
    atomicAdd(&cnt_out[g0], 1.0f);
    atomicAdd(&cnt_in[g1],  1.0f);
}

// ---------------- scatter-add with async global->LDS staging ----------------
__device__ __forceinline__ void async_stage_b128(uint32_t lds_off, const float* gp) {
    // NT: x is streamed exactly once; don't evict the L2-resident accumulator.
    asm volatile("global_load_async_to_lds_b128 %0, %1, off th:TH_LOAD_NT"
                 :: "v"(lds_off), "v"(gp)
                 : "memory");
}

__device__ __forceinline__ void do_atomics(float4 v, long chunk,
                                           const int2* __restrict__ gidx,
                                           float* __restrict__ out) {
    long e   = chunk / COLS4;
    int  c4  = (int)(chunk % COLS4);
    int  col = c4 * 4;                       // float4 never straddles col 256
    int2 g   = gidx[e];
    int  node = (col < SPLIT_COL) ? g.y : g.x;
    float* dst = out + (long)node * D_DIM + col;
    atomicAdd(dst + 0, v.x);
    atomicAdd(dst + 1, v.y);
    atomicAdd(dst + 2, v.z);
    atomicAdd(dst + 3, v.w);
}

__global__ void scatter_add_kernel(const float* __restrict__ x,
                                   const int2* __restrict__ gidx,
                                   float* __restrict__ out,
                                   long nchunks) {
    __shared__ float4 tile[STAGE_IT * 256];   // 48 KB of the 320 KB WGP LDS
    const int  tid  = threadIdx.x;
    const long base = (long)blockIdx.x * CHUNKS_PB;

    // Flat address of a __shared__ object is {aperture_hi32, lds_offset_lo32};
    // the async instruction's VDST wants the 32-bit LDS byte offset.
    const uint32_t lds_base = (uint32_t)(uintptr_t)&tile[0];

    if (base + CHUNKS_PB <= nchunks) {
        // -------- interior fast path: no guards, every wave issues all 12 ----
        #pragma unroll
        for (int i = 0; i < STAGE_IT; ++i) {
            uint32_t lo = lds_base + (uint32_t)(i * 256 + tid) * 16u;
            async_stage_b128(lo, x + (base + (long)i * 256 + tid) * 4);
        }
        // Consume chunk i as soon as transfer i lands; i+1..11 stay in flight.
        #pragma unroll
        for (int i = 0; i < STAGE_IT; ++i) {
            asm volatile("s_wait_asynccnt %0" :: "n"(STAGE_IT - 1 - i) : "memory");
            float4 v = tile[i * 256 + tid];          // ds_load_b128
            do_atomics(v, base + (long)i * 256 + tid, gidx, out);
        }
    } else {
        // -------- tail path: guarded issue, full drain -----------------------
        #pragma unroll
        for (int i = 0; i < STAGE_IT; ++i) {
            long chunk = base + (long)i * 256 + tid;
            if (chunk < nchunks) {
                uint32_t lo = lds_base + (uint32_t)(i * 256 + tid) * 16u;
                async_stage_b128(lo, x + chunk * 4);
            }
        }
        asm volatile("s_wait_asynccnt 0x0" ::: "memory");
        #pragma unroll
        for (int i = 0; i < STAGE_IT; ++i) {
            long chunk = base + (long)i * 256 + tid;
            if (chunk >= nchunks) continue;
            float4 v = tile[i * 256 + tid];
            do_atomics(v, chunk, gidx, out);
        }
    }
}

// ---------------- in-place normalize: sum / max(count, 1) ----------------
__global__ void normalize_kernel(float4* __restrict__ out4,
                                 const float* __restrict__ cnt_in,
                                 const float* __restrict__ cnt_out,
                                 long n4) {
    long i = (long)blockIdx.x * blockDim.x + threadIdx.x;
    if (i >= n4) return;
    long node = i / COLS4;
    int  c4   = (int)(i % COLS4);
    float c   = (c4 * 4 < SPLIT_COL) ? cnt_in[node] : cnt_out[node];
    float s   = 1.0f / fmaxf(c, 1.0f);
    float4 v  = out4[i];
    v.x *= s; v.y *= s; v.z *= s; v.w *= s;
    out4[i] = v;
}

// ---------------------------------------------------------------------------
static inline size_t align16(size_t v) { return (v + 15) & ~(size_t)15; }

extern "C" void kernel_launch(void* const* d_in, const int* in_sizes, int n_in,
                              void* d_out, int out_size, void* d_ws, size_t ws_size,
                              hipStream_t stream) {
    const float* x   = (const float*)d_in[0];   // [E, 384] f32
    const int*   gsz = (const int*)d_in[1];     // [B] i32
    const int*   ptr = (const int*)d_in[2];     // [B+1] i32
    const int*   lg  = (const int*)d_in[3];     // [E, 2] i32
    float*       out = (float*)d_out;           // [TN, 384] f32

    const int  B  = in_sizes[1];
    const long E  = (long)in_sizes[3] / 2;
    const long TN = (long)out_size / D_DIM;

    // workspace carve-out
    char*  w       = (char*)d_ws;
    float* cnt_in  = (float*)w;  w += align16((size_t)TN * sizeof(float));
    float* cnt_out = (float*)w;  w += align16((size_t)TN * sizeof(float));
    int2*  gidx    = (int2*)w;   w += align16((size_t)E * sizeof(int2));
    int*   offs    = (int*)w;
    (void)ws_size; (void)n_in;

    // 1) zero accumulator (output) and counts
    {
        long n4 = (long)out_size / 4;
        long nb = (n4 + 255) / 256;
        zero_f4_kernel<<<(unsigned)nb, 256, 0, stream>>>((float4*)out, n4);
        long nc = 2 * TN;
        long cb = (nc + 255) / 256;
        zero_f_kernel<<<(unsigned)cb, 256, 0, stream>>>(cnt_in, nc); // cnt_in & cnt_out contiguous
    }

    // 2) per-graph node offsets (cumsum of org_graph_size)
    offsets_kernel<<<1, 32, 0, stream>>>(gsz, B, offs);

    // 3) global endpoint indices + per-endpoint counts
    {
        long nb = (E + 255) / 256;
        gidx_count_kernel<<<(unsigned)nb, 256, 0, stream>>>(lg, ptr, offs, B, E,
                                                            gidx, cnt_in, cnt_out);
    }

    // 4) scatter-add of x into out (async LDS staging + f32 atomics)
    {
        long nchunks = E * COLS4;
        long nb = (nchunks + CHUNKS_PB - 1) / CHUNKS_PB;
        scatter_add_kernel<<<(unsigned)nb, 256, 0, stream>>>(x, gidx, out, nchunks);
    }

    // 5) normalize sums into means, in place
    {
        long n4 = (long)out_size / 4;
        long nb = (n4 + 255) / 256;
        normalize_kernel<<<(unsigned)nb, 256, 0, stream>>>((float4*)out,
                                                           cnt_in, cnt_out, n4);
    }
}